// DualModeHead_42245298323784
// MI455X (gfx1250) — compile-verified
//
#include <hip/hip_runtime.h>
#include <hip/hip_bf16.h>
#include <math.h>

typedef __attribute__((ext_vector_type(2))) float v2f;
typedef __attribute__((ext_vector_type(8))) float v8f;

#define B_  8
#define T_  2048
#define D_  128
#define WPB 4   // waves per block

// ---------------------------------------------------------------------------
// Pre-pass: Keff[b,s,d] = scale * ( (1-alpha)*k[b,s,(d-1)%D] - alpha*k[b,s,d] )
// alpha = sigmoid(mode). Folds meson/baryon mix + scale into one K matrix so
// the attention core is a single GEMM per score tile.
// ---------------------------------------------------------------------------
__global__ void dmh_keff_kernel(const float* __restrict__ k,
                                const float* __restrict__ mode,
                                float* __restrict__ keff, int n) {
  int idx = blockIdx.x * blockDim.x + threadIdx.x;
  if (idx >= n) return;
  float alpha = 1.0f / (1.0f + __expf(-mode[0]));
  const float scale = 0.08838834764831845f;  // 128^-0.5
  int d    = idx & (D_ - 1);
  int base = idx - d;
  float kc = k[idx];
  float kp = k[base + ((d + D_ - 1) & (D_ - 1))];
  keff[idx] = scale * ((1.0f - alpha) * kp - alpha * kc);
}

// ---------------------------------------------------------------------------
// Flash attention core. One wave owns a 16-row query tile and streams causal
// 16-key tiles, using V_WMMA_F32_16X16X4_F32 for both S = Q*Keff^T and O += P*V.
// Operand fetches are batched into register arrays so loads clause together
// and overlap the wmma chains instead of serializing on s_wait_loadcnt.
// ---------------------------------------------------------------------------
__global__ __launch_bounds__(WPB * 32)
void dmh_fattn_kernel(const float* __restrict__ q,
                      const float* __restrict__ keff,
                      const float* __restrict__ v,
                      float* __restrict__ out) {
  // per-wave P staging buffer, 16 rows x 16 cols, row stride 18 floats
  // (bank-conflict-free for both access patterns; DS ops in-order per wave)
  __shared__ float lds_p[WPB][16 * 18];

  const int lane = threadIdx.x & 31;
  const int wid  = threadIdx.x >> 5;
  const int h    = lane >> 4;   // lane-half (K-chunk select in A/B layouts)
  const int lp   = lane & 15;   // row (A) / col (B,C,D) within tile

  const int gwave = blockIdx.x * WPB + wid;   // 0..1023
  const int b  = gwave >> 7;                  // batch
  const int mt = gwave & 127;                 // M tile index
  const int m0 = mt << 4;

  // ---- preload Q tile in A-operand layout:
  // lane holds row (m0+lp), dims 4t + 2h + {0,1} for t = 0..31
  v2f qr[32];
  {
    const float* qrow = q + ((size_t)(b * T_ + m0 + lp)) * D_ + 2 * h;
#pragma unroll
    for (int t = 0; t < 32; ++t)
      qr[t] = *(const v2f*)(qrow + 4 * t);
  }

  // O accumulators: 8 dim-groups of 16 cols, C/D layout (row 8h+j in elem j)
  v8f o[8];
#pragma unroll
  for (int g = 0; g < 8; ++g)
#pragma unroll
    for (int j = 0; j < 8; ++j) o[g][j] = 0.0f;

  float mrow[8], lrow[8];
#pragma unroll
  for (int j = 0; j < 8; ++j) { mrow[j] = -INFINITY; lrow[j] = 0.0f; }

  for (int nt = 0; nt <= mt; ++nt) {
    const int n0 = nt << 4;

    // ---- S = Qtile @ Keff^T : 32 wmma over D=128, two accumulator chains
    v8f s0, s1;
#pragma unroll
    for (int j = 0; j < 8; ++j) { s0[j] = 0.0f; s1[j] = 0.0f; }
    const float* krow = keff + ((size_t)(b * T_ + n0 + lp)) * D_ + 2 * h;
#pragma unroll
    for (int c = 0; c < 4; ++c) {           // 4 chunks of 8 wmma
      v2f kbv[8];                           // batch loads -> one clause/wait
#pragma unroll
      for (int i = 0; i < 8; ++i)
        kbv[i] = *(const v2f*)(krow + 32 * c + 4 * i);
#pragma unroll
      for (int i = 0; i < 8; ++i) {
        if ((i & 1) == 0)
          s0 = __builtin_amdgcn_wmma_f32_16x16x4_f32(
              false, qr[8 * c + i], false, kbv[i], (short)0, s0, false, false);
        else
          s1 = __builtin_amdgcn_wmma_f32_16x16x4_f32(
              false, qr[8 * c + i], false, kbv[i], (short)0, s1, false, false);
      }
    }
    v8f s = s0 + s1;

    // ---- causal mask on the diagonal tile: key n0+lp vs row m0+8h+j
    if (nt == mt) {
#pragma unroll
      for (int j = 0; j < 8; ++j)
        if (lp > 8 * h + j) s[j] = -INFINITY;
    }

    // ---- online softmax (rows live across the 16 lanes of each half)
    float pj[8], rs[8], mnew[8];
#pragma unroll
    for (int j = 0; j < 8; ++j) {
      float vmax = s[j];
#pragma unroll
      for (int off = 1; off < 16; off <<= 1)
        vmax = fmaxf(vmax, __shfl_xor(vmax, off, 32));
      mnew[j] = fmaxf(mrow[j], vmax);
      pj[j]   = __expf(s[j] - mnew[j]);
      float rsum = pj[j];
#pragma unroll
      for (int off = 1; off < 16; off <<= 1)
        rsum += __shfl_xor(rsum, off, 32);
      rs[j] = rsum;
    }

    // stage P through LDS to convert C-layout -> A-layout
#pragma unroll
    for (int j = 0; j < 8; ++j)
      lds_p[wid][(8 * h + j) * 18 + lp] = pj[j];

    // rescale running O and stats
#pragma unroll
    for (int j = 0; j < 8; ++j) {
      float esc = __expf(mrow[j] - mnew[j]);
      lrow[j] = lrow[j] * esc + rs[j];
      mrow[j] = mnew[j];
#pragma unroll
      for (int g = 0; g < 8; ++g) o[g][j] *= esc;
    }

    // A operands for P@V: lane = row lp, keys 4kb + 2h + {0,1} (batched)
    v2f pa[4];
#pragma unroll
    for (int kb = 0; kb < 4; ++kb)
      pa[kb] = *(const v2f*)&lds_p[wid][lp * 18 + 4 * kb + 2 * h];

    // ---- O += P @ Vtile : 4 key sub-blocks x 8 dim-groups = 32 wmma
    const float* vbase = v + ((size_t)(b * T_ + n0)) * D_ + lp;
#pragma unroll
    for (int kb = 0; kb < 4; ++kb) {
      const float* vk0 = vbase + (size_t)(4 * kb + 2 * h) * D_;
      v2f vb[8];                            // batch 16 loads -> clause
#pragma unroll
      for (int g = 0; g < 8; ++g) {
        vb[g].x = vk0[16 * g];        // key 4kb+2h+0, dim 16g+lp
        vb[g].y = vk0[16 * g + D_];   // key 4kb+2h+1, dim 16g+lp
      }
#pragma unroll
      for (int g = 0; g < 8; ++g)
        o[g] = __builtin_amdgcn_wmma_f32_16x16x4_f32(
            false, pa[kb], false, vb[g], (short)0, o[g], false, false);
    }
  }

  // ---- epilogue: out = O / l
  float* obase = out + ((size_t)(b * T_ + m0 + 8 * h)) * D_ + lp;
#pragma unroll
  for (int j = 0; j < 8; ++j) {
    float inv = 1.0f / lrow[j];
#pragma unroll
    for (int g = 0; g < 8; ++g)
      obase[(size_t)j * D_ + 16 * g] = o[g][j] * inv;
  }
}

// ---------------------------------------------------------------------------
extern "C" void kernel_launch(void* const* d_in, const int* in_sizes, int n_in,
                              void* d_out, int out_size, void* d_ws, size_t ws_size,
                              hipStream_t stream) {
  const float* q    = (const float*)d_in[0];
  const float* k    = (const float*)d_in[1];
  const float* v    = (const float*)d_in[2];
  // d_in[3] = mask (causality handled analytically)
  const float* mode = (const float*)d_in[4];
  float* out  = (float*)d_out;
  float* keff = (float*)d_ws;   // needs B*T*D*4 = 8 MB of workspace

  const int n = B_ * T_ * D_;
  dmh_keff_kernel<<<(n + 255) / 256, 256, 0, stream>>>(k, mode, keff, n);

  const int total_waves = B_ * (T_ / 16);        // 1024
  const int blocks = total_waves / WPB;          // 256
  dmh_fattn_kernel<<<blocks, WPB * 32, 0, stream>>>(q, keff, v, out);
}